// GraphConvLayer_11536282157551
// MI455X (gfx1250) — compile-verified
//
#include <hip/hip_runtime.h>
#include <math.h>

typedef __attribute__((ext_vector_type(2))) float v2f;
typedef __attribute__((ext_vector_type(8))) float v8f;

#define NODE_F   64
#define EDGE_F   32
#define IN_DIM   160
#define KC_TOT   40          // 160 / 4
#define FRAG_PER_MAT (KC_TOT * 4 /*nt*/ * 32 /*lane*/ * 2 /*regs*/)   // 10240 floats
#define BN_EPS   1e-5f

__device__ __forceinline__ v8f wmma4(v2f a, v2f b, v8f c) {
    // D = A(16x4 f32) * B(4x16 f32) + C(16x16 f32)
    return __builtin_amdgcn_wmma_f32_16x16x4_f32(
        /*neg_a=*/false, a, /*neg_b=*/false, b,
        /*c_mod=*/(short)0, c, /*reuse_a=*/false, /*reuse_b=*/false);
}

__device__ __forceinline__ float sigmoid_f(float x) {
    return 1.0f / (1.0f + __expf(-x));
}
__device__ __forceinline__ float softplus_f(float x) {
    // numerically stable: max(x,0) + log1p(exp(-|x|))
    return fmaxf(x, 0.0f) + log1pf(__expf(-fabsf(x)));
}

// ---------------------------------------------------------------- zero init
__global__ void zero_ws_kernel(float* __restrict__ p, long long count) {
    long long i = (long long)blockIdx.x * blockDim.x + threadIdx.x;
    if (i < count) p[i] = 0.0f;
}

// ------------------------------------------------- edge GEMM + gate + scatter
__global__ __launch_bounds__(256) void edge_gemm_scatter(
    const float* __restrict__ node_attrs,
    const int*   __restrict__ edge_index,   // [2][E] int32
    const float* __restrict__ edge_attrs,
    const float* __restrict__ W_f, const float* __restrict__ b_f,
    const float* __restrict__ W_s, const float* __restrict__ b_s,
    float* __restrict__ msg,
    int nEdges, int nTiles, int wavesTotal)
{
    __shared__ float wlds[2 * FRAG_PER_MAT];   // 80 KB

    const int tid = threadIdx.x;

    // Stage W_f / W_s into LDS in B-fragment order:
    //   j = kc*256 + nt*64 + lane*2 + r
    //   element = W[kc*4 + (lane<16 ? r : 2+r)][nt*16 + (lane&15)]
    for (int i = tid; i < 2 * FRAG_PER_MAT; i += 256) {
        const int mat  = (i >= FRAG_PER_MAT) ? 1 : 0;
        const int j    = i - mat * FRAG_PER_MAT;
        const int r    = j & 1;
        const int ln   = (j >> 1) & 31;
        const int nt   = (j >> 6) & 3;
        const int kc   = j >> 8;
        const int kk   = (ln < 16) ? r : (2 + r);
        const int k    = kc * 4 + kk;
        const int n    = nt * 16 + (ln & 15);
        const float* W = mat ? W_s : W_f;
        wlds[i] = W[k * NODE_F + n];
    }
    __syncthreads();

    const int lane  = tid & 31;
    const int wave  = tid >> 5;
    const int gwave = blockIdx.x * 8 + wave;
    const int mrow  = lane & 15;              // A-matrix row handled by this lane
    const int koff  = (lane < 16) ? 0 : 2;    // A-matrix K offset for this half-wave
    const int half8 = (lane < 16) ? 0 : 8;    // C-matrix row offset for this half-wave

    const int* __restrict__ srcIdx = edge_index;
    const int* __restrict__ tgtIdx = edge_index + nEdges;

    for (int tile = gwave; tile < nTiles; tile += wavesTotal) {
        const int e  = tile * 16 + mrow;
        const int si = srcIdx[e];
        const int ti = tgtIdx[e];
        const float* pi = node_attrs + (long long)si * NODE_F + koff;  // k in [0,64)
        const float* pj = node_attrs + (long long)ti * NODE_F + koff;  // k in [64,128)
        const float* pe = edge_attrs + (long long)e  * EDGE_F + koff;  // k in [128,160)

        v8f accf[4] = {};   // sigmoid branch, 4 N-tiles of 16 cols
        v8f accs[4] = {};   // softplus branch

        // K-chunks 0..15 : x_i
        for (int kc = 0; kc < 16; ++kc) {
            v2f a = *(const v2f*)(pi + kc * 4);
            const float* wb0 = wlds + kc * 256 + lane * 2;
            const float* wb1 = wb0 + FRAG_PER_MAT;
#pragma unroll
            for (int nt = 0; nt < 4; ++nt)
                accf[nt] = wmma4(a, *(const v2f*)(wb0 + nt * 64), accf[nt]);
#pragma unroll
            for (int nt = 0; nt < 4; ++nt)
                accs[nt] = wmma4(a, *(const v2f*)(wb1 + nt * 64), accs[nt]);
        }
        // K-chunks 16..31 : x_j
        for (int kc = 16; kc < 32; ++kc) {
            v2f a = *(const v2f*)(pj + (kc - 16) * 4);
            const float* wb0 = wlds + kc * 256 + lane * 2;
            const float* wb1 = wb0 + FRAG_PER_MAT;
#pragma unroll
            for (int nt = 0; nt < 4; ++nt)
                accf[nt] = wmma4(a, *(const v2f*)(wb0 + nt * 64), accf[nt]);
#pragma unroll
            for (int nt = 0; nt < 4; ++nt)
                accs[nt] = wmma4(a, *(const v2f*)(wb1 + nt * 64), accs[nt]);
        }
        // K-chunks 32..39 : edge_attrs
        for (int kc = 32; kc < 40; ++kc) {
            v2f a = *(const v2f*)(pe + (kc - 32) * 4);
            const float* wb0 = wlds + kc * 256 + lane * 2;
            const float* wb1 = wb0 + FRAG_PER_MAT;
#pragma unroll
            for (int nt = 0; nt < 4; ++nt)
                accf[nt] = wmma4(a, *(const v2f*)(wb0 + nt * 64), accf[nt]);
#pragma unroll
            for (int nt = 0; nt < 4; ++nt)
                accs[nt] = wmma4(a, *(const v2f*)(wb1 + nt * 64), accs[nt]);
        }

        // Epilogue: bias + sigmoid*softplus, scatter-add into msg[src][col]
#pragma unroll
        for (int nt = 0; nt < 4; ++nt) {
            const int col = nt * 16 + mrow;
            const float bfv = b_f[col];
            const float bsv = b_s[col];
#pragma unroll
            for (int r = 0; r < 8; ++r) {
                const int er = tile * 16 + r + half8;   // edge row for C-reg r
                const float h = sigmoid_f(accf[nt][r] + bfv) *
                                softplus_f(accs[nt][r] + bsv);
                unsafeAtomicAdd(&msg[(long long)srcIdx[er] * NODE_F + col], h);
            }
        }
    }
}

// ----------------------------------------------- scalar tail (E % 16 edges)
__global__ void edge_tail_kernel(
    const float* __restrict__ node_attrs,
    const int*   __restrict__ edge_index,
    const float* __restrict__ edge_attrs,
    const float* __restrict__ W_f, const float* __restrict__ b_f,
    const float* __restrict__ W_s, const float* __restrict__ b_s,
    float* __restrict__ msg, int e0, int nEdges)
{
    const int e = e0 + blockIdx.x;
    if (e >= nEdges) return;
    const int col = threadIdx.x;   // 64 threads
    const int si = edge_index[e];
    const int ti = edge_index[nEdges + e];
    float af = b_f[col], as = b_s[col];
    for (int k = 0; k < NODE_F; ++k) {
        float z = node_attrs[(long long)si * NODE_F + k];
        af += z * W_f[k * NODE_F + col];
        as += z * W_s[k * NODE_F + col];
    }
    for (int k = 0; k < NODE_F; ++k) {
        float z = node_attrs[(long long)ti * NODE_F + k];
        af += z * W_f[(NODE_F + k) * NODE_F + col];
        as += z * W_s[(NODE_F + k) * NODE_F + col];
    }
    for (int k = 0; k < EDGE_F; ++k) {
        float z = edge_attrs[(long long)e * EDGE_F + k];
        af += z * W_f[(2 * NODE_F + k) * NODE_F + col];
        as += z * W_s[(2 * NODE_F + k) * NODE_F + col];
    }
    const float h = sigmoid_f(af) * softplus_f(as);
    unsafeAtomicAdd(&msg[(long long)si * NODE_F + col], h);
}

// ---------------------------------------------------- column stats (f64 acc)
__global__ __launch_bounds__(256) void col_stats_kernel(
    const float* __restrict__ msg, int nNodes, double* __restrict__ stats /*[128]*/)
{
    __shared__ double ssum[256];
    __shared__ double ssq[256];
    const int tid = threadIdx.x;
    const int col = tid & 63;
    const int rg  = tid >> 6;   // 0..3
    double s = 0.0, s2 = 0.0;
    for (int n = blockIdx.x * 4 + rg; n < nNodes; n += gridDim.x * 4) {
        const float v = msg[(long long)n * NODE_F + col];
        s  += (double)v;
        s2 += (double)v * (double)v;
    }
    ssum[tid] = s;
    ssq[tid]  = s2;
    __syncthreads();
    if (tid < 64) {
        const double ts = ssum[tid] + ssum[tid + 64] + ssum[tid + 128] + ssum[tid + 192];
        const double t2 = ssq[tid]  + ssq[tid + 64]  + ssq[tid + 128]  + ssq[tid + 192];
        unsafeAtomicAdd(&stats[col],      ts);
        unsafeAtomicAdd(&stats[64 + col], t2);
    }
}

// ------------------------------------------------- batchnorm + residual out
__global__ void bn_residual_kernel(
    const float* __restrict__ node_attrs,
    const float* __restrict__ msg,
    const double* __restrict__ stats,
    const float* __restrict__ gamma,
    const float* __restrict__ beta,
    float* __restrict__ out, int nNodes)
{
    const long long total = (long long)nNodes * NODE_F;
    const long long i = (long long)blockIdx.x * blockDim.x + threadIdx.x;
    if (i >= total) return;
    const int f = (int)(i & (NODE_F - 1));
    const double invN = 1.0 / (double)nNodes;
    const double mean = stats[f] * invN;
    double var = stats[64 + f] * invN - mean * mean;
    if (var < 0.0) var = 0.0;
    const float inv = rsqrtf((float)var + BN_EPS);
    const float normed = (msg[i] - (float)mean) * inv * gamma[f] + beta[f];
    out[i] = node_attrs[i] + normed;
}

// ---------------------------------------------------------------- launcher
extern "C" void kernel_launch(void* const* d_in, const int* in_sizes, int n_in,
                              void* d_out, int out_size, void* d_ws, size_t ws_size,
                              hipStream_t stream) {
    const float* node_attrs = (const float*)d_in[0];
    const int*   edge_index = (const int*)d_in[1];
    const float* edge_attrs = (const float*)d_in[2];
    const float* W_f   = (const float*)d_in[3];
    const float* b_f   = (const float*)d_in[4];
    const float* W_s   = (const float*)d_in[5];
    const float* b_s   = (const float*)d_in[6];
    const float* gamma = (const float*)d_in[7];
    const float* beta  = (const float*)d_in[8];

    const int nNodes = in_sizes[0] / NODE_F;
    const int nEdges = in_sizes[1] / 2;

    float*  msg   = (float*)d_ws;
    double* stats = (double*)((char*)d_ws + (size_t)nNodes * NODE_F * sizeof(float));

    // 1) zero message buffer + stats (128 doubles == 256 floats of zero bits)
    {
        const long long count = (long long)nNodes * NODE_F + 256;
        const int blocks = (int)((count + 255) / 256);
        zero_ws_kernel<<<blocks, 256, 0, stream>>>(msg, count);
    }

    // 2) edge GEMM (WMMA f32 16x16x4) + gate + scatter
    const int nTiles = nEdges / 16;
    const int rem    = nEdges - nTiles * 16;
    if (nTiles > 0) {
        int blocks = 1250;                               // 10000 waves
        const int maxBlocks = (nTiles + 7) / 8;
        if (blocks > maxBlocks) blocks = maxBlocks;
        const int wavesTotal = blocks * 8;
        edge_gemm_scatter<<<blocks, 256, 0, stream>>>(
            node_attrs, edge_index, edge_attrs, W_f, b_f, W_s, b_s,
            msg, nEdges, nTiles, wavesTotal);
    }
    if (rem > 0) {
        edge_tail_kernel<<<rem, 64, 0, stream>>>(
            node_attrs, edge_index, edge_attrs, W_f, b_f, W_s, b_s,
            msg, nTiles * 16, nEdges);
    }

    // 3) per-column mean / sumsq in double
    col_stats_kernel<<<512, 256, 0, stream>>>(msg, nNodes, stats);

    // 4) batchnorm + residual
    {
        const long long total = (long long)nNodes * NODE_F;
        const int blocks = (int)((total + 255) / 256);
        bn_residual_kernel<<<blocks, 256, 0, stream>>>(
            node_attrs, msg, stats, gamma, beta, (float*)d_out, nNodes);
    }
}